// fusion_loss_med3D_90056874263104
// MI455X (gfx1250) — compile-verified
//
#include <hip/hip_runtime.h>
#include <hip/hip_bf16.h>
#include <math.h>

// Problem: 128^3 fp32 volumes. Workspace use: 7 * 2M floats + 4 doubles ~= 56MB.
#define DIM   128
#define NVOX  (128*128*128)
#define C1_   1.0e-4f
#define C2_   9.0e-4f

typedef float v8f __attribute__((ext_vector_type(8)));
typedef float v2f __attribute__((ext_vector_type(2)));

// Normalized 11-tap Gaussian (sigma=1.5): g[d] = exp(-(d-5)^2/4.5)/S, 1/S = g[5].
__device__ __forceinline__ float gaussw(int d) {
    float x = (float)(d - 5);
    return __expf(x * x * (-1.0f / 4.5f)) * 0.26601172f;
}

// ---------------------------------------------------------------------------
// Separable 11-tap Gaussian pass along one axis, one field.
// MODE: 0 -> field = in1 ; 1 -> field = in1*in1 ; 2 -> field = in1*in2
// SC = conv-axis stride, SA/SB = strides of the two line axes (all compile-time).
// Each wave computes a 16(pos) x 16(line) output tile as
//   D(16x16) = sum_{kt=0..6} A_kt(16x4) * B_kt(4x16)      (zero-padded band)
// using V_WMMA_F32_16X16X4_F32. Fragment layouts per CDNA5 ISA 7.12.2:
//   A: lane L holds M=L&15, K = {2*(L>>4), 2*(L>>4)+1} within the K-tile
//   B: lane L holds N=L&15, same K split
//   D: lane L, vgpr v holds M = v + 8*(L>>4), N = L&15
// Zero padding is branchless: clamp the address, load, cndmask-select 0.
// ---------------------------------------------------------------------------
template<int MODE, int SC, int SA, int SB>
__global__ __launch_bounds__(256) void gauss_pass(
    const float* __restrict__ in1, const float* __restrict__ in2,
    float* __restrict__ out)
{
    const int lane  = threadIdx.x & 31;
    const int tile  = blockIdx.x * 8 + (threadIdx.x >> 5);  // 8192 tiles total
    const int posT  = tile & 7;                              // 8 pos-tiles / line
    const int lineT = tile >> 3;                             // 1024 line groups
    const int P0    = posT << 4;
    const int nLo   = lane & 15;
    const int hi    = lane >> 4;
    const int r     = (lineT << 4) | nLo;                    // line id 0..16383
    const int base  = (r & 127) * SA + (r >> 7) * SB;

    // Branchless padded load of the field value at conv position ip.
    auto loadB = [&](int ip) -> float {
        int ipc = ip < 0 ? 0 : (ip > 127 ? 127 : ip);        // always in-bounds
        int idx = base + ipc * SC;
        float v = in1[idx];
        if (MODE == 1)      v = v * v;
        else if (MODE == 2) v = v * in2[idx];
        return ((unsigned)ip < 128u) ? v : 0.0f;             // v_cndmask
    };

    v8f c = {0.f, 0.f, 0.f, 0.f, 0.f, 0.f, 0.f, 0.f};

#if __has_builtin(__builtin_amdgcn_wmma_f32_16x16x4_f32)
    const int khalf = hi << 1;
    const int m = nLo;
    #pragma unroll
    for (int kt = 0; kt < 7; ++kt) {
        const int q0 = (kt << 2) + khalf;
        v2f a, b;
        int d0 = q0 - m;                                     // band offset
        a.x = ((unsigned)d0       <= 10u) ? gaussw(d0)     : 0.0f;
        a.y = ((unsigned)(d0 + 1) <= 10u) ? gaussw(d0 + 1) : 0.0f;
        b.x = loadB(P0 - 5 + q0);
        b.y = loadB(P0 - 4 + q0);
        c = __builtin_amdgcn_wmma_f32_16x16x4_f32(
                false, a, false, b, (short)0, c, false, false);
    }
#else
    // Correct scalar fallback with identical D fragment mapping.
    #pragma unroll
    for (int v = 0; v < 8; ++v) {
        int pos = P0 + v + (hi << 3);
        float s = 0.f;
        for (int j = 0; j < 11; ++j) s += gaussw(j) * loadB(pos - 5 + j);
        c[v] = s;
    }
#endif

    #pragma unroll
    for (int v = 0; v < 8; ++v) {
        int pos = P0 + v + (hi << 3);
        out[base + pos * SC] = c[v];
    }
}

// ---------------------------------------------------------------------------
// Block reduction (wave32 shuffle + LDS) -> f64 global atomic accumulate.
// ---------------------------------------------------------------------------
__device__ __forceinline__ void reduce_add(float v, int tid, double* acc) {
    for (int off = 16; off > 0; off >>= 1) v += __shfl_down(v, off, 32);
    __shared__ float sred[8];
    if ((tid & 31) == 0) sred[tid >> 5] = v;
    __syncthreads();
    if (tid == 0) {
        float s = 0.f;
        #pragma unroll
        for (int i = 0; i < 8; ++i) s += sred[i];
        atomicAdd(acc, (double)s);
    }
}

// ---------------------------------------------------------------------------
// L1 intensity term: sum |F - max(A,B)|
// ---------------------------------------------------------------------------
__global__ __launch_bounds__(256) void l1_kernel(
    const float* __restrict__ A, const float* __restrict__ B,
    const float* __restrict__ F, double* acc)
{
    float s = 0.f;
    for (int i = blockIdx.x * 256 + threadIdx.x; i < NVOX;
         i += gridDim.x * 256)
        s += fabsf(F[i] - fmaxf(A[i], B[i]));
    reduce_add(s, threadIdx.x, acc);
}

// ---------------------------------------------------------------------------
// All six Sobel-gradient terms in one LDS-tiled kernel.
// Orientations: (h,w) plane @ fixed d ; (d,w) @ fixed h ; (d,h) @ fixed w.
// sum += |gF - max(gA,gB)|^2 for sobel_x and sobel_y in each plane.
// ---------------------------------------------------------------------------
#define SOBEL(T, FETCH, gx, gy) {                                    \
    float v00=FETCH(T,0,0), v01=FETCH(T,0,1), v02=FETCH(T,0,2);      \
    float v10=FETCH(T,1,0),                  v12=FETCH(T,1,2);       \
    float v20=FETCH(T,2,0), v21=FETCH(T,2,1), v22=FETCH(T,2,2);      \
    gx = (v02 - v00) + 2.0f*(v12 - v10) + (v22 - v20);               \
    gy = (v00 + 2.0f*v01 + v02) - (v20 + 2.0f*v21 + v22); }

#define F_P1(T,i,j) T[z][y-1+i][x-1+j]
#define F_P2(T,i,j) T[z-1+i][y][x-1+j]
#define F_P3(T,i,j) T[z-1+i][y-1+j][x]

__global__ __launch_bounds__(256) void grad_kernel(
    const float* __restrict__ A, const float* __restrict__ B,
    const float* __restrict__ F, double* acc)
{
    __shared__ float tA[6][10][10], tB[6][10][10], tF[6][10][10];
    const int bx = blockIdx.x * 8, by = blockIdx.y * 8, bz = blockIdx.z * 4;
    const int tid = threadIdx.z * 64 + threadIdx.y * 8 + threadIdx.x;

    for (int i = tid; i < 600; i += 256) {
        int x = i % 10, y = (i / 10) % 10, z = i / 100;
        int gx = bx + x - 1, gy = by + y - 1, gz = bz + z - 1;
        bool ok = (unsigned)gx < 128u && (unsigned)gy < 128u && (unsigned)gz < 128u;
        int idx = ok ? (gz * 16384 + gy * 128 + gx) : 0;
        tA[z][y][x] = ok ? A[idx] : 0.f;
        tB[z][y][x] = ok ? B[idx] : 0.f;
        tF[z][y][x] = ok ? F[idx] : 0.f;
    }
    __syncthreads();

    const int x = threadIdx.x + 1, y = threadIdx.y + 1, z = threadIdx.z + 1;
    float s = 0.f;
    float axv, ayv, bxv, byv, fxv, fyv, t;

    SOBEL(tA, F_P1, axv, ayv); SOBEL(tB, F_P1, bxv, byv); SOBEL(tF, F_P1, fxv, fyv);
    t = fxv - fmaxf(axv, bxv); s += t * t;
    t = fyv - fmaxf(ayv, byv); s += t * t;

    SOBEL(tA, F_P2, axv, ayv); SOBEL(tB, F_P2, bxv, byv); SOBEL(tF, F_P2, fxv, fyv);
    t = fxv - fmaxf(axv, bxv); s += t * t;
    t = fyv - fmaxf(ayv, byv); s += t * t;

    SOBEL(tA, F_P3, axv, ayv); SOBEL(tB, F_P3, bxv, byv); SOBEL(tF, F_P3, fxv, fyv);
    t = fxv - fmaxf(axv, bxv); s += t * t;
    t = fyv - fmaxf(ayv, byv); s += t * t;

    reduce_add(s, tid, acc);
}

// ---------------------------------------------------------------------------
// SSIM map for one pair (X, F) from 5 blurred fields, reduced to a sum.
// ---------------------------------------------------------------------------
__global__ __launch_bounds__(256) void ssim_reduce(
    const float* __restrict__ GX,  const float* __restrict__ GF,
    const float* __restrict__ GX2, const float* __restrict__ GF2,
    const float* __restrict__ GXF, double* acc)
{
    float s = 0.f;
    for (int i = blockIdx.x * 256 + threadIdx.x; i < NVOX;
         i += gridDim.x * 256) {
        float mu1 = GX[i], mu2 = GF[i];
        float m11 = mu1 * mu1, m22 = mu2 * mu2, m12 = mu1 * mu2;
        float s1 = GX2[i] - m11, s2 = GF2[i] - m22, s12 = GXF[i] - m12;
        float num = (2.f * m12 + C1_) * (2.f * s12 + C2_);
        float den = (m11 + m22 + C1_) * (s1 + s2 + C2_);
        s += num / den;
    }
    reduce_add(s, threadIdx.x, acc);
}

__global__ void zero_acc(double* acc) {
    if (threadIdx.x < 4) acc[threadIdx.x] = 0.0;
}

__global__ void finalize_kernel(const double* __restrict__ acc,
                                float* __restrict__ out)
{
    const double n = (double)NVOX;
    double l1   = 2.0  * fabs(acc[0] / n);
    double grad = 10.0 * (acc[1] / n);
    double sa   = fabs(acc[2] / n);
    double sb   = fabs(acc[3] / n);
    double sl   = fabs(0.5 * (1.0 - sa) + 0.5 * (1.0 - sb));
    out[0] = (float)(l1 + grad + sl);   // fusion_loss
    out[1] = (float)grad;               // loss_gradient
    out[2] = (float)l1;                 // loss_l1
    out[3] = (float)sl;                 // loss_SSIM
}

// ---------------------------------------------------------------------------
extern "C" void kernel_launch(void* const* d_in, const int* in_sizes, int n_in,
                              void* d_out, int out_size, void* d_ws, size_t ws_size,
                              hipStream_t stream)
{
    const float* A = (const float*)d_in[0];
    const float* B = (const float*)d_in[1];
    const float* F = (const float*)d_in[2];
    float* out = (float*)d_out;

    float* ws     = (float*)d_ws;
    float* slotF  = ws + 0 * (size_t)NVOX;   // G*F
    float* slotF2 = ws + 1 * (size_t)NVOX;   // G*(F*F)
    float* slotX  = ws + 2 * (size_t)NVOX;   // G*X
    float* slotX2 = ws + 3 * (size_t)NVOX;   // G*(X*X)
    float* slotXF = ws + 4 * (size_t)NVOX;   // G*(X*F)
    float* tmpA   = ws + 5 * (size_t)NVOX;
    float* tmpB   = ws + 6 * (size_t)NVOX;
    double* acc   = (double*)(ws + 7 * (size_t)NVOX);   // 4 accumulators

    zero_acc<<<1, 32, 0, stream>>>(acc);
    l1_kernel<<<1024, 256, 0, stream>>>(A, B, F, acc + 0);
    grad_kernel<<<dim3(16, 16, 32), dim3(8, 8, 4), 0, stream>>>(A, B, F, acc + 1);

    // Separable Gaussian blur: z pass (strides 16384/1/128), then y, then x.
    auto conv3 = [&](const float* s1, const float* s2, int mode, float* dst) {
        if (mode == 0)
            gauss_pass<0, 16384, 1, 128><<<1024, 256, 0, stream>>>(s1, s2, tmpA);
        else if (mode == 1)
            gauss_pass<1, 16384, 1, 128><<<1024, 256, 0, stream>>>(s1, s2, tmpA);
        else
            gauss_pass<2, 16384, 1, 128><<<1024, 256, 0, stream>>>(s1, s2, tmpA);
        gauss_pass<0, 128, 1, 16384><<<1024, 256, 0, stream>>>(tmpA, nullptr, tmpB);
        gauss_pass<0, 1, 128, 16384><<<1024, 256, 0, stream>>>(tmpB, nullptr, dst);
    };

    conv3(F, nullptr, 0, slotF);
    conv3(F, nullptr, 1, slotF2);

    conv3(A, nullptr, 0, slotX);
    conv3(A, nullptr, 1, slotX2);
    conv3(A, F,       2, slotXF);
    ssim_reduce<<<1024, 256, 0, stream>>>(slotX, slotF, slotX2, slotF2, slotXF, acc + 2);

    conv3(B, nullptr, 0, slotX);
    conv3(B, nullptr, 1, slotX2);
    conv3(B, F,       2, slotXF);
    ssim_reduce<<<1024, 256, 0, stream>>>(slotX, slotF, slotX2, slotF2, slotXF, acc + 3);

    finalize_kernel<<<1, 1, 0, stream>>>(acc, out);

    (void)in_sizes; (void)n_in; (void)out_size; (void)ws_size;
}